// ICP_10033043603806
// MI455X (gfx1250) — compile-verified
//
#include <hip/hip_runtime.h>
#include <math.h>

typedef __attribute__((ext_vector_type(2))) float v2f;
typedef __attribute__((ext_vector_type(8))) float v8f;

#define NB 8
#define NP 2048
#define STEPLIM 10
#define TOLF 1e-4f
#define EPSF 1e-12f

// workspace layout (in 4-byte units)
#define PC_OFF   0                       // NB*NP*3 floats (current point cloud)
#define SUMS_OFF (NB*NP*3)               // NB*16 floats: sp[3], sq[3], S[9], sumdist
#define RT_OFF   (SUMS_OFF + NB*16)      // NB*12 floats: R[9], t[3]
#define ERR_OFF  (RT_OFF + NB*12)        // NB floats
#define DONE_OFF (ERR_OFF + NB)          // 1 int

// ---------------------------------------------------------------- init
__global__ void icp_init(const float* __restrict__ psrc, float* __restrict__ ws) {
    int i = blockIdx.x * blockDim.x + threadIdx.x;
    if (i < NB * NP * 3) ws[PC_OFF + i] = psrc[i];
    if (i < NB) ws[ERR_OFF + i] = 0.0f;
    if (i == 0) ((int*)ws)[DONE_OFF] = 0;
}

// ------------------------------------------------- NN search + reduction
// One block per batch. WMMA f32 16x16x4 computes 16x16 dot-product tiles
// (K=3 padded to 4). score = |b|^2 - 2 a.b ; |a|^2 added later (argmin-safe).
// LDS target array stores (x,y,z,|b|^2) so the inner loop is one ds_load_b128
// + one v_wmma + min/argmin selects.
__global__ __launch_bounds__(256) void icp_nn(const float* __restrict__ ptgt,
                                              float* __restrict__ ws) {
    __shared__ float4 stgt[NP];      // 32 KB, w = |t|^2
    __shared__ float  sScore[NP];    //  8 KB
    __shared__ int    sIdx[NP];      //  8 KB
    __shared__ float  red[256];      //  1 KB

    const int b   = blockIdx.x;
    const int tid = threadIdx.x;
    const float* tg = ptgt + b * NP * 3;
    const float* pc = ws + PC_OFF + b * NP * 3;

    for (int i = tid; i < NP; i += 256) {
        float x = tg[i*3+0], y = tg[i*3+1], z = tg[i*3+2];
        stgt[i] = make_float4(x, y, z, x*x + y*y + z*z);
    }
    __syncthreads();

    const int lane  = tid & 31;
    const int wave  = tid >> 5;
    const int l16   = lane & 15;
    const int khalf = lane >> 4;

    for (int mt = wave; mt < NP / 16; mt += 8) {
        const int m0   = mt * 16;
        const int mrow = m0 + l16;
        const float px = pc[mrow*3+0], py = pc[mrow*3+1], pz = pc[mrow*3+2];
        v2f a;
        a.x = khalf ? pz : px;       // K=0 (lo half) / K=2 (hi half)
        a.y = khalf ? 0.0f : py;     // K=1 (lo half) / K=3 (hi half, zero pad)

        float best[8]; int bidx[8];
        #pragma unroll
        for (int r = 0; r < 8; ++r) { best[r] = 3.4e38f; bidx[r] = 0; }

        #pragma unroll 2
        for (int n0 = 0; n0 < NP; n0 += 16) {
            float4 t = stgt[n0 + l16];
            v2f bb;
            bb.x = khalf ? t.z : t.x;
            bb.y = khalf ? 0.0f : t.y;
            v8f c = {};
            c = __builtin_amdgcn_wmma_f32_16x16x4_f32(
                    false, a, false, bb, (short)0, c, false, false);
            const int cand = n0 + l16;
            #pragma unroll
            for (int r = 0; r < 8; ++r) {
                float s = fmaf(-2.0f, c[r], t.w);
                if (s < best[r]) { best[r] = s; bidx[r] = cand; }
            }
        }
        // min/argmin across the 16 lanes of each half (rows stay separate)
        #pragma unroll
        for (int r = 0; r < 8; ++r) {
            float s = best[r]; int id = bidx[r];
            #pragma unroll
            for (int off = 1; off < 16; off <<= 1) {
                float os = __shfl_xor(s, off, 32);
                int   oi = __shfl_xor(id, off, 32);
                if (os < s || (os == s && oi < id)) { s = os; id = oi; }
            }
            best[r] = s; bidx[r] = id;
        }
        if (l16 == 0) {  // lane 0 -> rows m0..m0+7, lane 16 -> rows m0+8..m0+15
            #pragma unroll
            for (int r = 0; r < 8; ++r) {
                int row = m0 + khalf * 8 + r;
                sScore[row] = best[r];
                sIdx[row]   = bidx[r];
            }
        }
    }
    __syncthreads();

    // per-thread partial sums: sp[3], sq[3], S[9] (raw cross moments), sumdist
    float acc[16];
    #pragma unroll
    for (int k = 0; k < 16; ++k) acc[k] = 0.0f;
    for (int m = tid; m < NP; m += 256) {
        float px = pc[m*3+0], py = pc[m*3+1], pz = pc[m*3+2];
        float a2 = px*px + py*py + pz*pz;
        float d2 = fmaxf(a2 + sScore[m], EPSF);
        float4 q = stgt[sIdx[m]];
        acc[0] += px; acc[1] += py; acc[2] += pz;
        acc[3] += q.x; acc[4] += q.y; acc[5] += q.z;
        acc[6]  += px*q.x; acc[7]  += px*q.y; acc[8]  += px*q.z;
        acc[9]  += py*q.x; acc[10] += py*q.y; acc[11] += py*q.z;
        acc[12] += pz*q.x; acc[13] += pz*q.y; acc[14] += pz*q.z;
        acc[15] += sqrtf(d2);
    }
    float* sums = ws + SUMS_OFF + b * 16;
    for (int k = 0; k < 16; ++k) {
        red[tid] = acc[k];
        __syncthreads();
        for (int s = 128; s > 0; s >>= 1) {
            if (tid < s) red[tid] += red[tid + s];
            __syncthreads();
        }
        if (tid == 0) sums[k] = red[0];
        __syncthreads();
    }
}

// --------------------------------------------- Kabsch via Horn quaternion
__device__ void horn_kabsch(const float* s, float* R, float* t) {
    const float invN = 1.0f / (float)NP;
    float cs0 = s[0]*invN, cs1 = s[1]*invN, cs2 = s[2]*invN;
    float ct0 = s[3]*invN, ct1 = s[4]*invN, ct2 = s[5]*invN;
    float H[9];
    #pragma unroll
    for (int i = 0; i < 3; ++i)
        #pragma unroll
        for (int j = 0; j < 3; ++j)
            H[i*3+j] = s[6 + i*3 + j] - s[i] * s[3 + j] * invN;
    float Sxx=H[0],Sxy=H[1],Sxz=H[2],Syx=H[3],Syy=H[4],Syz=H[5],Szx=H[6],Szy=H[7],Szz=H[8];
    float Nq[16];
    Nq[0]=Sxx+Syy+Szz; Nq[1]=Syz-Szy;      Nq[2]=Szx-Sxz;       Nq[3]=Sxy-Syx;
    Nq[4]=Nq[1];       Nq[5]=Sxx-Syy-Szz;  Nq[6]=Sxy+Syx;       Nq[7]=Szx+Sxz;
    Nq[8]=Nq[2];       Nq[9]=Nq[6];        Nq[10]=-Sxx+Syy-Szz; Nq[11]=Syz+Szy;
    Nq[12]=Nq[3];      Nq[13]=Nq[7];       Nq[14]=Nq[11];       Nq[15]=-Sxx-Syy+Szz;
    float shift = 1.0f;
    #pragma unroll
    for (int i = 0; i < 16; ++i) shift += fabsf(Nq[i]);
    float v0 = 1.0f, v1 = 0.01f, v2 = 0.02f, v3 = 0.03f;
    for (int it = 0; it < 60; ++it) {
        float w0 = shift*v0 + Nq[0]*v0  + Nq[1]*v1  + Nq[2]*v2  + Nq[3]*v3;
        float w1 = shift*v1 + Nq[4]*v0  + Nq[5]*v1  + Nq[6]*v2  + Nq[7]*v3;
        float w2 = shift*v2 + Nq[8]*v0  + Nq[9]*v1  + Nq[10]*v2 + Nq[11]*v3;
        float w3 = shift*v3 + Nq[12]*v0 + Nq[13]*v1 + Nq[14]*v2 + Nq[15]*v3;
        float nrm = rsqrtf(w0*w0 + w1*w1 + w2*w2 + w3*w3 + 1e-30f);
        v0 = w0*nrm; v1 = w1*nrm; v2 = w2*nrm; v3 = w3*nrm;
    }
    float qw=v0, qx=v1, qy=v2, qz=v3;
    R[0]=1.0f-2.0f*(qy*qy+qz*qz); R[1]=2.0f*(qx*qy-qw*qz);      R[2]=2.0f*(qx*qz+qw*qy);
    R[3]=2.0f*(qx*qy+qw*qz);      R[4]=1.0f-2.0f*(qx*qx+qz*qz); R[5]=2.0f*(qy*qz-qw*qx);
    R[6]=2.0f*(qx*qz-qw*qy);      R[7]=2.0f*(qy*qz+qw*qx);      R[8]=1.0f-2.0f*(qx*qx+qy*qy);
    t[0] = ct0 - (R[0]*cs0 + R[1]*cs1 + R[2]*cs2);
    t[1] = ct1 - (R[3]*cs0 + R[4]*cs1 + R[5]*cs2);
    t[2] = ct2 - (R[6]*cs0 + R[7]*cs1 + R[8]*cs2);
}

// one wave; lane b < NB handles batch b; global done across all batches
__global__ void icp_step(float* __restrict__ ws) {
    __shared__ int convs[NB];
    const int lane = threadIdx.x;
    int* donep = (int*)ws + DONE_OFF;
    const int done_old = *donep;
    if (lane < NB) {
        const float* s = ws + SUMS_OFF + lane * 16;
        float errnew = s[15] / (float)NP;
        float errold = ws[ERR_OFF + lane];
        convs[lane] = (fabsf(errnew - errold) < TOLF) ? 1 : 0;
    }
    __syncthreads();
    int all = 1;
    for (int i = 0; i < NB; ++i) all &= convs[i];
    const int done_new = done_old | all;
    if (lane < NB) {
        const float* s = ws + SUMS_OFF + lane * 16;
        float R[9], t[3];
        horn_kabsch(s, R, t);
        float* rt = ws + RT_OFF + lane * 12;
        #pragma unroll
        for (int k = 0; k < 9; ++k) rt[k] = R[k];
        rt[9] = t[0]; rt[10] = t[1]; rt[11] = t[2];
        float errnew = s[15] / (float)NP;
        if (!done_new) ws[ERR_OFF + lane] = errnew;
    }
    if (lane == 0) *donep = done_new;
}

// pc = R pc + t (skipped once globally converged)
__global__ void icp_apply(float* __restrict__ ws) {
    int i = blockIdx.x * blockDim.x + threadIdx.x;
    if (i >= NB * NP) return;
    if (((const int*)ws)[DONE_OFF]) return;
    int b = i / NP;
    float* pc = ws + PC_OFF;
    float px = pc[i*3+0], py = pc[i*3+1], pz = pc[i*3+2];
    const float* rt = ws + RT_OFF + b * 12;
    float nx = rt[0]*px + rt[1]*py + rt[2]*pz + rt[9];
    float ny = rt[3]*px + rt[4]*py + rt[5]*pz + rt[10];
    float nz = rt[6]*px + rt[7]*py + rt[8]*pz + rt[11];
    pc[i*3+0] = nx; pc[i*3+1] = ny; pc[i*3+2] = nz;
}

// matched-pair sums for the final Kabsch(psrc, pc_final)
__global__ __launch_bounds__(256) void icp_pairsum(const float* __restrict__ p,
                                                   const float* __restrict__ q,
                                                   float* __restrict__ sums) {
    __shared__ float red[256];
    const int b = blockIdx.x, tid = threadIdx.x;
    const float* pb = p + b * NP * 3;
    const float* qb = q + b * NP * 3;
    float acc[16];
    #pragma unroll
    for (int k = 0; k < 16; ++k) acc[k] = 0.0f;
    for (int m = tid; m < NP; m += 256) {
        float px = pb[m*3+0], py = pb[m*3+1], pz = pb[m*3+2];
        float qx = qb[m*3+0], qy = qb[m*3+1], qz = qb[m*3+2];
        acc[0] += px; acc[1] += py; acc[2] += pz;
        acc[3] += qx; acc[4] += qy; acc[5] += qz;
        acc[6]  += px*qx; acc[7]  += px*qy; acc[8]  += px*qz;
        acc[9]  += py*qx; acc[10] += py*qy; acc[11] += py*qz;
        acc[12] += pz*qx; acc[13] += pz*qy; acc[14] += pz*qz;
    }
    for (int k = 0; k < 16; ++k) {
        red[tid] = acc[k];
        __syncthreads();
        for (int s = 128; s > 0; s >>= 1) {
            if (tid < s) red[tid] += red[tid + s];
            __syncthreads();
        }
        if (tid == 0) sums[b * 16 + k] = red[0];
        __syncthreads();
    }
}

__global__ void icp_final(const float* __restrict__ ws, float* __restrict__ out) {
    const int lane = threadIdx.x;
    if (lane < NB) {
        const float* s = ws + SUMS_OFF + lane * 16;
        float R[9], t[3];
        horn_kabsch(s, R, t);
        float* o = out + lane * 12;  // [3,4] row-major: [R | t]
        #pragma unroll
        for (int i = 0; i < 3; ++i) {
            o[i*4+0] = R[i*3+0];
            o[i*4+1] = R[i*3+1];
            o[i*4+2] = R[i*3+2];
            o[i*4+3] = t[i];
        }
    }
}

extern "C" void kernel_launch(void* const* d_in, const int* in_sizes, int n_in,
                              void* d_out, int out_size, void* d_ws, size_t ws_size,
                              hipStream_t stream) {
    const float* psrc = (const float*)d_in[0];
    const float* ptgt = (const float*)d_in[1];
    float* ws  = (float*)d_ws;
    float* out = (float*)d_out;

    icp_init<<<(NB*NP*3 + 255) / 256, 256, 0, stream>>>(psrc, ws);
    for (int it = 0; it < STEPLIM; ++it) {
        icp_nn<<<NB, 256, 0, stream>>>(ptgt, ws);
        icp_step<<<1, 32, 0, stream>>>(ws);
        icp_apply<<<(NB*NP + 255) / 256, 256, 0, stream>>>(ws);
    }
    icp_pairsum<<<NB, 256, 0, stream>>>(psrc, ws + PC_OFF, ws + SUMS_OFF);
    icp_final<<<1, 32, 0, stream>>>(ws, out);
}